// QLSTM_65481071398712
// MI455X (gfx1250) — compile-verified
//
#include <hip/hip_runtime.h>
#include <hip/hip_bf16.h>

typedef float v2f __attribute__((ext_vector_type(2)));
typedef float v8f __attribute__((ext_vector_type(8)));

namespace {
constexpr int B_ = 256, T_ = 32, D_ = 256, H_ = 256;
constexpr int M_ = B_ * T_;             // 8192 rows of X
constexpr int TILES_ = M_ / 16;         // 512
constexpr int YS_N = B_ * T_ * H_;      // 2097152
constexpr int HX_N = B_ * H_;           // 65536
constexpr int OUT_N = YS_N + 2 * HX_N;  // 2228224
}

// ---------------------------------------------------------------------------
// Kernel 1: per-gate circuit constants (alpha, beta, gamma) + hidden-sum s_h.
// E(theta) = alpha + beta*cos(theta) + gamma*sin(theta).
// One block of 256 threads; thread k owns basis amplitude k of the 8-qubit state.
// ---------------------------------------------------------------------------
__device__ __forceinline__ float block_sum_256(float v, float* red, int tid) {
    red[tid] = v;
    __syncthreads();
#pragma unroll
    for (int s = 128; s > 0; s >>= 1) {
        if (tid < s) red[tid] += red[tid + s];
        __syncthreads();
    }
    float r = red[0];
    __syncthreads();
    return r;
}

__global__ __launch_bounds__(256)
void qlstm_consts(const float* __restrict__ rxf, const float* __restrict__ rzf,
                  const float* __restrict__ rxi, const float* __restrict__ rzi,
                  const float* __restrict__ rxg, const float* __restrict__ rzg,
                  const float* __restrict__ rxo, const float* __restrict__ rzo,
                  const float* __restrict__ Wf,  const float* __restrict__ Wi,
                  const float* __restrict__ Wg,  const float* __restrict__ Wo,
                  float* __restrict__ consts)   // [16]: alpha[4], beta[4], gamma[4], s_h[4]
{
    __shared__ float red[256];
    const int k = threadIdx.x;

    // CNOT-ladder permutation (wire w = bit 7-w; control bit 7..1 -> flip next lower bit)
    int p = k;
#pragma unroll
    for (int bit = 7; bit >= 1; --bit)
        if ((p >> bit) & 1) p ^= (1 << (bit - 1));
    const float z = (8.0f - 2.0f * (float)__popc(p)) * 0.125f;  // mean <Z_w> weight

    const float* rxs[4] = {rxf, rxi, rxg, rxo};
    const float* rzs[4] = {rzf, rzi, rzg, rzo};
    const float* Ws[4]  = {Wf, Wi, Wg, Wo};

    for (int gidx = 0; gidx < 4; ++gidx) {
        const float* rx = rxs[gidx];
        const float* rz = rzs[gidx];
        // u = U|e0>, v = U|e128> as product states (before CNOT permutation)
        float ur = 1.f, ui = 0.f, vr = 1.f, vi = 0.f;
#pragma unroll
        for (int w = 0; w < 8; ++w) {
            const int bit = (k >> (7 - w)) & 1;
            const float cxh = cosf(0.5f * rx[w]), sxh = sinf(0.5f * rx[w]);
            const float czh = cosf(0.5f * rz[w]), szh = sinf(0.5f * rz[w]);
            // G = RZ*RX;  G|0> components / G|1> components
            float g0r, g0i;
            if (bit == 0) { g0r =  czh * cxh; g0i = -szh * cxh; }
            else          { g0r =  szh * sxh; g0i = -czh * sxh; }
            float nur = ur * g0r - ui * g0i;
            float nui = ur * g0i + ui * g0r;
            ur = nur; ui = nui;
            float g1r = g0r, g1i = g0i;
            if (w == 0) {  // wire 0 of v uses G|1>
                if (bit == 0) { g1r = -szh * sxh; g1i = -czh * sxh; }
                else          { g1r =  czh * cxh; g1i =  szh * cxh; }
            }
            float nvr = vr * g1r - vi * g1i;
            float nvi = vr * g1i + vi * g1r;
            vr = nvr; vi = nvi;
        }
        const float A  = block_sum_256(z * (ur * ur + ui * ui), red, k);
        const float Bv = block_sum_256(z * (vr * vr + vi * vi), red, k);
        const float Q  = block_sum_256(z * (ur * vi - ui * vr), red, k);
        const float sh = block_sum_256(Ws[gidx][256 + k], red, k);  // sum of W[0, D:D+H]
        if (k == 0) {
            consts[gidx]      = 0.5f * (A + Bv);
            consts[4 + gidx]  = 0.5f * (A - Bv);
            consts[8 + gidx]  = Q;
            consts[12 + gidx] = sh;
        }
        __syncthreads();
    }
}

// ---------------------------------------------------------------------------
// Kernel 2: angx[m][g] = X[m,0:256] . W_g[0,0:256] + b_g[0] via f32 WMMA.
// One wave per 16-row tile; gates live in columns 0..3 of the 16x16 D tile.
// A layout (16x4 f32): lanes 0-15 hold K=0 (v0), K=1 (v1); lanes 16-31 K=2,3.
// B layout (4x16 f32): mirrored — lanes 0-15 rows K=0,1; lanes 16-31 rows K=2,3.
// ---------------------------------------------------------------------------
__global__ __launch_bounds__(32)
void qlstm_angx_wmma(const float* __restrict__ X,
                     const float* __restrict__ Wf, const float* __restrict__ Wi,
                     const float* __restrict__ Wg, const float* __restrict__ Wo,
                     const float* __restrict__ bf, const float* __restrict__ bi,
                     const float* __restrict__ bg, const float* __restrict__ bo,
                     float* __restrict__ angx)    // [M_, 4]
{
    const int tile = blockIdx.x;          // 0..511
    const int lane = threadIdx.x;         // 0..31
    const int half = lane >> 4;           // 0|1
    const int ln   = lane & 15;           // N index (and A row-in-tile)
    const int row  = tile * 16 + ln;      // A row this lane supplies
    const int g    = ln & 3;
    const float* wg = (g == 0) ? Wf : (g == 1) ? Wi : (g == 2) ? Wg : Wo;
    const float bmask = (ln < 4) ? 1.0f : 0.0f;   // columns >=4 of B are zero
    const float* xrow = X + (long)row * D_;
    __builtin_prefetch(xrow, 0, 0);               // global_prefetch_b8

    v8f acc = {0.f, 0.f, 0.f, 0.f, 0.f, 0.f, 0.f, 0.f};
    for (int kk = 0; kk < D_; kk += 4) {
        const int k0 = kk + 2 * half;
        v2f a, bv;
        a.x  = xrow[k0];
        a.y  = xrow[k0 + 1];
        bv.x = wg[k0] * bmask;
        bv.y = wg[k0 + 1] * bmask;
        acc = __builtin_amdgcn_wmma_f32_16x16x4_f32(false, a, false, bv,
                                                    (short)0, acc, false, false);
    }
    // D layout: VGPR r -> (M = r + 8*half, N = ln)
    if (ln < 4) {
        const float bias = (ln == 0) ? bf[0] : (ln == 1) ? bi[0] : (ln == 2) ? bg[0] : bo[0];
#pragma unroll
        for (int r = 0; r < 8; ++r) {
            const int m = tile * 16 + r + 8 * half;
            angx[m * 4 + ln] = acc[r] + bias;
        }
    }
}

// ---------------------------------------------------------------------------
// Kernel 3: the recurrent scan. One thread per batch element (state is rank-1).
// ---------------------------------------------------------------------------
__global__ __launch_bounds__(256)
void qlstm_scan(const float* __restrict__ angx, const float* __restrict__ consts,
                float* __restrict__ hscal,      // [M_] h scalar per (b,t)
                float* __restrict__ clast)      // [B_] final cell scalar
{
    const int b = threadIdx.x;
    float al[4], be[4], ga[4], sh[4];
#pragma unroll
    for (int q = 0; q < 4; ++q) {
        al[q] = consts[q];
        be[q] = consts[4 + q];
        ga[q] = consts[8 + q];
        sh[q] = consts[12 + q];
    }
    float h = 0.f, c = 0.f;
    for (int t = 0; t < T_; ++t) {
        const int m = b * T_ + t;
        const float4 ax = reinterpret_cast<const float4*>(angx)[m];
        const float axv[4] = {ax.x, ax.y, ax.z, ax.w};
        float q[4];
#pragma unroll
        for (int gq = 0; gq < 4; ++gq) {
            const float th = axv[gq] + h * sh[gq];
            q[gq] = al[gq] + be[gq] * cosf(th) + ga[gq] * sinf(th);
        }
        const float f  = 1.0f / (1.0f + expf(-q[0]));
        const float ii = 1.0f / (1.0f + expf(-q[1]));
        const float gg = tanhf(q[2]);
        const float oo = 1.0f / (1.0f + expf(-q[3]));
        c = f * c + ii * gg;
        h = oo * tanhf(c);
        hscal[m] = h;
    }
    clast[b] = c;
}

// ---------------------------------------------------------------------------
// Kernel 4: broadcast scalars into ys [B,T,H] | hx [B,H] | cx [B,H] (float4).
// ---------------------------------------------------------------------------
__global__ __launch_bounds__(256)
void qlstm_bcast(const float* __restrict__ hscal, const float* __restrict__ clast,
                 float* __restrict__ out)
{
    const int i = blockIdx.x * blockDim.x + threadIdx.x;  // float4 index
    constexpr int n4 = OUT_N / 4;
    if (i >= n4) return;
    const int e = i * 4;
    float val;
    if (e < YS_N) {
        val = hscal[e >> 8];                      // /H: same h across hidden dim
    } else if (e < YS_N + HX_N) {
        const int b = (e - YS_N) >> 8;
        val = hscal[b * T_ + (T_ - 1)];
    } else {
        const int b = (e - YS_N - HX_N) >> 8;
        val = clast[b];
    }
    float4 v; v.x = v.y = v.z = v.w = val;
    reinterpret_cast<float4*>(out)[i] = v;
}

// ---------------------------------------------------------------------------
extern "C" void kernel_launch(void* const* d_in, const int* in_sizes, int n_in,
                              void* d_out, int out_size, void* d_ws, size_t ws_size,
                              hipStream_t stream)
{
    // setup_inputs() dict order: inputs, then per gate n in [f,i,g,o]: W,b,rx,rz
    const float* X   = (const float*)d_in[0];
    const float* Wf  = (const float*)d_in[1];
    const float* bf  = (const float*)d_in[2];
    const float* rxf = (const float*)d_in[3];
    const float* rzf = (const float*)d_in[4];
    const float* Wi  = (const float*)d_in[5];
    const float* bi  = (const float*)d_in[6];
    const float* rxi = (const float*)d_in[7];
    const float* rzi = (const float*)d_in[8];
    const float* Wg  = (const float*)d_in[9];
    const float* bg  = (const float*)d_in[10];
    const float* rxg = (const float*)d_in[11];
    const float* rzg = (const float*)d_in[12];
    const float* Wo  = (const float*)d_in[13];
    const float* bo  = (const float*)d_in[14];
    const float* rxo = (const float*)d_in[15];
    const float* rzo = (const float*)d_in[16];

    float* ws     = (float*)d_ws;
    float* consts = ws;                   // 16 floats
    float* angx   = ws + 16;              // 8192*4 floats (16B-aligned)
    float* hscal  = angx + M_ * 4;        // 8192 floats
    float* clast  = hscal + M_;           // 256 floats

    qlstm_consts<<<1, 256, 0, stream>>>(rxf, rzf, rxi, rzi, rxg, rzg, rxo, rzo,
                                        Wf, Wi, Wg, Wo, consts);
    qlstm_angx_wmma<<<TILES_, 32, 0, stream>>>(X, Wf, Wi, Wg, Wo,
                                               bf, bi, bg, bo, angx);
    qlstm_scan<<<1, 256, 0, stream>>>(angx, consts, hscal, clast);
    constexpr int n4 = OUT_N / 4;
    qlstm_bcast<<<(n4 + 255) / 256, 256, 0, stream>>>(hscal, clast, (float*)d_out);
}